// CHROM_31533649887367
// MI455X (gfx1250) — compile-verified
//
#include <hip/hip_runtime.h>
#include <hip/hip_bf16.h>
#include <cmath>
#include <complex>

typedef __attribute__((ext_vector_type(2))) float v2f;
typedef __attribute__((ext_vector_type(4))) float vf4;
typedef __attribute__((ext_vector_type(8))) float v8f;

// ---------------------------------------------------------------------------
// Kernel 1: spatial mean over 128x128 per (b,c,t) slice.
// 9216 blocks x 256 threads; each thread streams 16 float4 (coalesced b128).
// Cross-lane reduction uses V_WMMA_F32_16X16X4_F32 (exact fp32 matrix op):
//   A[m][k] holds each lane's partial once (B = ones) -> D rows = lane sums.
// ---------------------------------------------------------------------------
__global__ __launch_bounds__(256) void spatial_mean_kernel(
    const float* __restrict__ x, float* __restrict__ means) {
  const int slice = blockIdx.x;          // b*144 + c*48 + t (input layout order)
  const int b = slice / 144;
  const int c = (slice / 48) % 3;
  const int t = slice % 48;

  const vf4* p = reinterpret_cast<const vf4*>(x) + (size_t)slice * 4096;
  float sum = 0.f;
  const int i = threadIdx.x;
#pragma unroll
  for (int k = 0; k < 16; ++k) {
    vf4 v = __builtin_nontemporal_load(&p[i + k * 256]);   // streaming: don't pollute L2
    sum += (v.x + v.y) + (v.z + v.w);
  }

  // Wave32 reduction via fp32 WMMA: D[m][n] = sum_k A[m][k] * 1.0
  v2f a;  a.x = sum; a.y = 0.f;          // lanes 0-15 -> A[m][0], lanes 16-31 -> A[m][2]
  v2f bo; bo.x = 1.f; bo.y = 1.f;        // B = all ones (layout-independent)
  v8f cacc = {};
  v8f d = __builtin_amdgcn_wmma_f32_16x16x4_f32(false, a, false, bo,
                                                (short)0, cacc, false, false);
  // D vgpr r: lane n<16 holds rowsum[r]; lane 16+n holds rowsum[r+8].
  __shared__ float s_red[16];
  const int lane = threadIdx.x & 31;
  const int wave = threadIdx.x >> 5;
  if (lane == 0 || lane == 16) {
    float loc = ((d[0] + d[1]) + (d[2] + d[3])) + ((d[4] + d[5]) + (d[6] + d[7]));
    s_red[wave * 2 + (lane >> 4)] = loc;
  }
  __syncthreads();
  if (threadIdx.x == 0) {
    float tot = 0.f;
#pragma unroll
    for (int k = 0; k < 16; ++k) tot += s_red[k];
    means[(size_t)(b * 48 + t) * 3 + c] = tot * (1.0f / 16384.0f);
  }
}

// ---------------------------------------------------------------------------
// Kernel 2: CHROM windows + filtfilt + overlap-add. One block, 128 threads,
// one thread per window (127 windows). Sequential IIR scans in LDS scratch.
// ---------------------------------------------------------------------------
struct FiltCoef { float b[7]; float a[7]; float zi[6]; };

__device__ inline void lfilter_fwd(float* x, const FiltCoef& fc) {
  const float x0 = x[0];
  float z0 = fc.zi[0] * x0, z1 = fc.zi[1] * x0, z2 = fc.zi[2] * x0;
  float z3 = fc.zi[3] * x0, z4 = fc.zi[4] * x0, z5 = fc.zi[5] * x0;
  for (int t = 0; t < 90; ++t) {
    float xt = x[t];
    float y = fc.b[0] * xt + z0;
    z0 = fc.b[1] * xt - fc.a[1] * y + z1;
    z1 = fc.b[2] * xt - fc.a[2] * y + z2;
    z2 = fc.b[3] * xt - fc.a[3] * y + z3;
    z3 = fc.b[4] * xt - fc.a[4] * y + z4;
    z4 = fc.b[5] * xt - fc.a[5] * y + z5;
    z5 = fc.b[6] * xt - fc.a[6] * y;
    x[t] = y;
  }
}

__device__ inline void lfilter_bwd(float* x, const FiltCoef& fc) {
  const float x0 = x[89];
  float z0 = fc.zi[0] * x0, z1 = fc.zi[1] * x0, z2 = fc.zi[2] * x0;
  float z3 = fc.zi[3] * x0, z4 = fc.zi[4] * x0, z5 = fc.zi[5] * x0;
  for (int t = 89; t >= 0; --t) {
    float xt = x[t];
    float y = fc.b[0] * xt + z0;
    z0 = fc.b[1] * xt - fc.a[1] * y + z1;
    z1 = fc.b[2] * xt - fc.a[2] * y + z2;
    z2 = fc.b[3] * xt - fc.a[3] * y + z3;
    z3 = fc.b[4] * xt - fc.a[4] * y + z4;
    z4 = fc.b[5] * xt - fc.a[5] * y + z5;
    z5 = fc.b[6] * xt - fc.a[6] * y;
    x[t] = y;
  }
}

__device__ inline void filtfilt_inplace(float* x, const FiltCoef& fc) {
  lfilter_fwd(x, fc);
  lfilter_bwd(x, fc);
}

__global__ __launch_bounds__(128) void chrom_kernel(
    const float* __restrict__ means, float* __restrict__ out, FiltCoef fc) {
  __shared__ float s_means[9216];      // [3072][3]
  __shared__ float s_ext[128 * 92];    // per-thread 90-sample filter scratch
  __shared__ float s_Xf[128 * 48];     // Xf then reused for S (window output)

  const int tid = threadIdx.x;
  for (int i = tid; i < 9216; i += 128) s_means[i] = means[i];
  __syncthreads();

  float* ext = &s_ext[tid * 92];
  float* Xf  = &s_Xf[tid * 48];
  const int w = tid;
  const bool active = (w < 127);

  if (active) {
    // per-window channel baselines
    float s0 = 0.f, s1 = 0.f;
    for (int t = 0; t < 48; ++t) {
      const float* m = &s_means[(w * 24 + t) * 3];
      s0 += m[0]; s1 += m[1];
    }
    const float base0 = s0 * (1.f / 48.f), base1 = s1 * (1.f / 48.f);

    // ---- X chrominance: 3*n0 - 2*n1 ----
    for (int t = 0; t < 48; ++t) {
      const float* m = &s_means[(w * 24 + t) * 3];
      ext[21 + t] = 3.f * (m[0] / base0) - 2.f * (m[1] / base1);
    }
    for (int i = 0; i < 21; ++i) ext[i]      = 2.f * ext[21] - ext[42 - i];  // odd left pad
    for (int i = 0; i < 21; ++i) ext[69 + i] = 2.f * ext[68] - ext[67 - i];  // odd right pad
    filtfilt_inplace(ext, fc);
    for (int t = 0; t < 48; ++t) Xf[t] = ext[21 + t];

    // ---- Y chrominance: 1.5*n0 - 0.5*n1 (as written in source) ----
    for (int t = 0; t < 48; ++t) {
      const float* m = &s_means[(w * 24 + t) * 3];
      ext[21 + t] = 1.5f * (m[0] / base0) - 0.5f * (m[1] / base1);
    }
    for (int i = 0; i < 21; ++i) ext[i]      = 2.f * ext[21] - ext[42 - i];
    for (int i = 0; i < 21; ++i) ext[69 + i] = 2.f * ext[68] - ext[67 - i];
    filtfilt_inplace(ext, fc);

    // std ratio (ddof=1 cancels in the ratio)
    float mx = 0.f, my = 0.f;
    for (int t = 0; t < 48; ++t) { mx += Xf[t]; my += ext[21 + t]; }
    mx *= (1.f / 48.f); my *= (1.f / 48.f);
    float vx = 0.f, vy = 0.f;
    for (int t = 0; t < 48; ++t) {
      float dx = Xf[t] - mx;       vx += dx * dx;
      float dy = ext[21 + t] - my; vy += dy * dy;
    }
    const float ratio = sqrtf(vx / vy);

    // S = (Xf - ratio*Yf) * hann ; store S over Xf
    for (int t = 0; t < 48; ++t) {
      float hann = 0.5f - 0.5f * cosf(6.283185307179586f * (float)t / 47.f);
      Xf[t] = (Xf[t] - ratio * ext[21 + t]) * hann;
    }
  }
  __syncthreads();

  // overlap-add: seg[s][j] = S[s][j] (s<127) + S[s-1][24+j] (s>=1)
  for (int j = 0; j < 24; ++j) {
    float v = 0.f;
    if (tid < 127) v += s_Xf[tid * 48 + j];
    if (tid >= 1)  v += s_Xf[(tid - 1) * 48 + 24 + j];
    out[tid * 24 + j] = v;
  }
}

// ---------------------------------------------------------------------------
// Host-side replication of scipy butter(3, bandpass) + lfilter_zi (float64,
// cast to f32 like the reference). Pure CPU math, deterministic.
// ---------------------------------------------------------------------------
static void solve6(double M[6][6], double Bv[6], double zi[6]) {
  for (int col = 0; col < 6; ++col) {
    int pr = col; double best = std::fabs(M[col][col]);
    for (int r = col + 1; r < 6; ++r)
      if (std::fabs(M[r][col]) > best) { best = std::fabs(M[r][col]); pr = r; }
    if (pr != col) {
      for (int k = 0; k < 6; ++k) { double tmp = M[col][k]; M[col][k] = M[pr][k]; M[pr][k] = tmp; }
      double tb = Bv[col]; Bv[col] = Bv[pr]; Bv[pr] = tb;
    }
    for (int r = col + 1; r < 6; ++r) {
      double f = M[r][col] / M[col][col];
      for (int k = col; k < 6; ++k) M[r][k] -= f * M[col][k];
      Bv[r] -= f * Bv[col];
    }
  }
  for (int r = 5; r >= 0; --r) {
    double s = Bv[r];
    for (int k = r + 1; k < 6; ++k) s -= M[r][k] * zi[k];
    zi[r] = s / M[r][r];
  }
}

static void compute_filter(FiltCoef* fc) {
  typedef std::complex<double> C;
  const double PI = 3.14159265358979323846;
  C p[3];
  for (int i = 0; i < 3; ++i) {
    int m = -2 + 2 * i;                      // -2, 0, 2
    p[i] = -std::exp(C(0.0, PI * m / 6.0));
  }
  const double wn0 = 0.7 / 36.0, wn1 = 2.5 / 36.0, fs = 2.0;
  const double w0 = 2 * fs * std::tan(PI * wn0 / fs);
  const double w1 = 2 * fs * std::tan(PI * wn1 / fs);
  const double bw = w1 - w0, wo = std::sqrt(w0 * w1);
  C pbp[6];
  for (int i = 0; i < 3; ++i) {
    C plp = p[i] * (bw * 0.5);
    C s = std::sqrt(plp * plp - C(wo * wo, 0.0));
    pbp[i] = plp + s; pbp[3 + i] = plp - s;
  }
  const double kbp = bw * bw * bw, fs2 = 2 * fs;
  C pd[6], prod(1.0, 0.0);
  for (int i = 0; i < 6; ++i) {
    pd[i] = (C(fs2, 0.0) + pbp[i]) / (C(fs2, 0.0) - pbp[i]);
    prod *= (C(fs2, 0.0) - pbp[i]);
  }
  const double zd[6] = {1, 1, 1, -1, -1, -1};
  const double kd = kbp * (C(fs2 * fs2 * fs2, 0.0) / prod).real();
  double bc[7] = {1, 0, 0, 0, 0, 0, 0};
  for (int i = 0; i < 6; ++i)
    for (int j = 6; j >= 1; --j) bc[j] -= zd[i] * bc[j - 1];
  C ac[7]; ac[0] = 1.0; for (int i = 1; i < 7; ++i) ac[i] = 0.0;
  for (int i = 0; i < 6; ++i)
    for (int j = 6; j >= 1; --j) ac[j] = ac[j] - pd[i] * ac[j - 1];
  double bd[7], ad[7];
  for (int i = 0; i < 7; ++i) { bd[i] = kd * bc[i]; ad[i] = ac[i].real(); }
  // lfilter_zi: solve (I - comp^T) zi = b[1:] - a[1:]*b[0]
  double comp[6][6] = {};
  for (int j = 0; j < 6; ++j) comp[0][j] = -ad[j + 1] / ad[0];
  for (int i = 1; i < 6; ++i) comp[i][i - 1] = 1.0;
  double M[6][6], Bv[6], zi[6];
  for (int i = 0; i < 6; ++i)
    for (int j = 0; j < 6; ++j) M[i][j] = (i == j ? 1.0 : 0.0) - comp[j][i];
  for (int j = 0; j < 6; ++j) Bv[j] = bd[j + 1] - ad[j + 1] * bd[0];
  solve6(M, Bv, zi);
  for (int i = 0; i < 7; ++i) { fc->b[i] = (float)bd[i]; fc->a[i] = (float)ad[i]; }
  for (int i = 0; i < 6; ++i) fc->zi[i] = (float)zi[i];
}

extern "C" void kernel_launch(void* const* d_in, const int* in_sizes, int n_in,
                              void* d_out, int out_size, void* d_ws, size_t ws_size,
                              hipStream_t stream) {
  (void)in_sizes; (void)n_in; (void)out_size; (void)ws_size;
  const float* x = (const float*)d_in[0];       // [64,3,48,128,128] f32
  float* out = (float*)d_out;                   // [64,48] f32
  float* means = (float*)d_ws;                  // [3072,3] f32 scratch

  FiltCoef fc;
  compute_filter(&fc);

  spatial_mean_kernel<<<9216, 256, 0, stream>>>(x, means);
  chrom_kernel<<<1, 128, 0, stream>>>(means, out, fc);
}